// CrossAttention_S_25701084299776
// MI455X (gfx1250) — compile-verified
//
#include <hip/hip_runtime.h>
#include <hip/hip_bf16.h>

typedef __attribute__((ext_vector_type(2))) float v2f;
typedef __attribute__((ext_vector_type(8))) float v8f;

#define CLASS_NUM 16
#define EMBED_DIM 64
#define NTOK      256
#define PAIRS     (CLASS_NUM * EMBED_DIM)   // 1024
#define ROWS      (PAIRS * NTOK / EMBED_DIM) // 4096 rows of the (4096,64) matrix

// ---------------------------------------------------------------------------
// Phase 1: rank-1 softmax attention per (c,e) pair.
//   o[i] = sum_j exp(q_i * k_j * 0.125) * v_j / sum_j exp(q_i * k_j * 0.125)
// One block per (c,e) pair, one thread per token i, k/v staged in LDS
// (uniform-j reads are LDS broadcasts, no bank conflicts).
// Storing o[c,e,n] contiguously yields exactly the (4096,64) row-major matrix
// the reference produces via reshape(-1, E).
// ---------------------------------------------------------------------------
__global__ __launch_bounds__(NTOK)
void attn_rank1_softmax_kernel(const float* __restrict__ q,
                               const float* __restrict__ k,
                               const float* __restrict__ v,
                               float* __restrict__ o) {
    __shared__ float sk[NTOK];
    __shared__ float sv[NTOK];

    const int pair = blockIdx.x;     // c*EMBED_DIM + e
    const int i    = threadIdx.x;    // token index 0..255
    const size_t base = (size_t)pair * NTOK;

    sk[i] = k[base + i];
    sv[i] = v[base + i];
    __syncthreads();

    const float s = q[base + i] * 0.125f;   // 1/sqrt(64)

    float num = 0.0f;
    float den = 0.0f;
    #pragma unroll 4
    for (int j = 0; j < NTOK; ++j) {
        const float e = __expf(s * sk[j]);  // v_exp_f32 (TRANS pipe)
        num = __fmaf_rn(e, sv[j], num);
        den += e;
    }
    o[base + i] = num / den;
}

// ---------------------------------------------------------------------------
// Phase 2: out(4096x64) = O(4096x64) @ W_out^T(64x64) + b, via
// V_WMMA_F32_16X16X4_F32 (fp32 A/B, f32 accum; 8-arg builtin form).
//   final[r,d] = sum_c O[r,c] * W[d,c] + b[d]
//   A[M=r,K=c] = O[r,c]          B[K=c,N=d] = W[d,c]
// One wave handles a 16-row tile across all four 16-col tiles (K=64 ->
// 16 WMMA steps x 4 tiles). Bias pre-loaded into the C accumulator.
//
// Fragment layouts (ISA 7.12.2, 32-bit):
//   A 16x4 : lane&15 = M ; lane>=16 -> K in {2,3} else {0,1} ; .x/.y = K pair
//   B 4x16 : lane&15 = N ; same K split as A
//   C/D    : VGPR g -> row g + 8*(lane>=16), col = lane&15
// ---------------------------------------------------------------------------
__global__ __launch_bounds__(256)
void out_proj_wmma_kernel(const float* __restrict__ O,
                          const float* __restrict__ W,     // W_out, (64,64) row-major: W[d*64+c]
                          const float* __restrict__ bias,  // b_out, (64)
                          float* __restrict__ out) {
    const int lane  = threadIdx.x & 31;
    const int wave  = threadIdx.x >> 5;          // 8 waves per block
    const int row0  = (blockIdx.x * 8 + wave) * 16;
    const int m     = lane & 15;                 // M for A, N for B/C/D
    const int khalf = lane >> 4;                 // 0: K={0,1}, 1: K={2,3}

    v8f acc[4];
    #pragma unroll
    for (int nt = 0; nt < 4; ++nt) {
        const float bv = bias[nt * 16 + m];      // column is fixed per lane
        v8f t = {bv, bv, bv, bv, bv, bv, bv, bv};
        acc[nt] = t;
    }

    const float* arow = O + (size_t)(row0 + m) * EMBED_DIM + khalf * 2;

    #pragma unroll
    for (int kk = 0; kk < 16; ++kk) {            // K = 64 in steps of 4
        const int kb = kk * 4;
        const v2f a = *(const v2f*)(arow + kb);  // A[m, kb + 2*khalf + {0,1}]
        #pragma unroll
        for (int nt = 0; nt < 4; ++nt) {
            const int d = nt * 16 + m;
            const v2f b = *(const v2f*)(W + (size_t)d * EMBED_DIM + kb + khalf * 2);
            acc[nt] = __builtin_amdgcn_wmma_f32_16x16x4_f32(
                /*neg_a=*/false, a, /*neg_b=*/false, b,
                /*c_mod=*/(short)0, acc[nt],
                /*reuse_a=*/false, /*reuse_b=*/false);
        }
    }

    #pragma unroll
    for (int nt = 0; nt < 4; ++nt) {
        const int col = nt * 16 + m;
        #pragma unroll
        for (int g = 0; g < 8; ++g) {
            const int row = row0 + g + khalf * 8;
            out[(size_t)row * EMBED_DIM + col] = acc[nt][g];
        }
    }
}

extern "C" void kernel_launch(void* const* d_in, const int* in_sizes, int n_in,
                              void* d_out, int out_size, void* d_ws, size_t ws_size,
                              hipStream_t stream) {
    (void)in_sizes; (void)n_in; (void)out_size; (void)ws_size;

    const float* q    = (const float*)d_in[0];
    const float* kk   = (const float*)d_in[1];
    const float* v    = (const float*)d_in[2];
    const float* Wout = (const float*)d_in[3];
    const float* bout = (const float*)d_in[4];
    float* out = (float*)d_out;

    float* O = (float*)d_ws;  // 4096 x 64 floats = 1 MB attention output

    attn_rank1_softmax_kernel<<<PAIRS, NTOK, 0, stream>>>(q, kk, v, O);
    out_proj_wmma_kernel<<<ROWS / 16 / 8, 256, 0, stream>>>(O, Wout, bout, out);
}